// PredictIntervention2Model_42494406427552
// MI455X (gfx1250) — compile-verified
//
#include <hip/hip_runtime.h>
#include <math.h>

typedef __attribute__((ext_vector_type(2))) float v2f;
typedef __attribute__((ext_vector_type(8))) float v8f;
typedef __attribute__((ext_vector_type(4))) unsigned int u32x4;
typedef __attribute__((ext_vector_type(4))) int i32x4;
typedef __attribute__((ext_vector_type(8))) int i32x8;

#define SEQLEN 96
#define EDIM   768
#define HDIM   768
#define GDIM   3072   /* 4*H */
#define NCONV  95

#if __has_builtin(__builtin_amdgcn_wmma_f32_16x16x4_f32)
#define HAVE_WMMA_F32 1
__device__ __forceinline__ v8f wmma_f32(v2f a, v2f b, v8f c) {
  return __builtin_amdgcn_wmma_f32_16x16x4_f32(false, a, false, b, (short)0, c, false, false);
}
#endif

#if __has_builtin(__builtin_amdgcn_tensor_load_to_lds)
#define HAVE_TDM 1
#endif

__device__ __forceinline__ float sigmoidf_(float x) {
  return 1.0f / (1.0f + __expf(-x));
}

// ---------------------------------------------------------------------------
// Stage the full activation matrix X (768x96 f32, 288 KB) into LDS.
// Preferred path: one TDM TENSOR_LOAD_TO_LDS issued by wave 0 with a 2D D#
// built per CDNA5 ISA, then s_wait_tensorcnt 0.
// Fallback: cooperative copy.
// ---------------------------------------------------------------------------
__device__ __forceinline__ void stage_x_lds(const float* __restrict__ X,
                                            float* lds_x) {
#ifdef HAVE_TDM
  if (threadIdx.x < 32) {          // one wave issues the DMA (EXEC ignored by TDM)
    unsigned lds_addr = (unsigned)(uintptr_t)lds_x;          // low 32b = LDS byte offset
    unsigned long long ga = (unsigned long long)(uintptr_t)X;
    u32x4 g0;
    g0[0] = 1u;                                              // count=1, user D#
    g0[1] = lds_addr;                                        // bits[63:32] lds_addr
    g0[2] = (unsigned)(ga & 0xFFFFFFFFu);                    // global_addr[31:0]
    g0[3] = (unsigned)((ga >> 32) & 0x1FFFFFFu) | (2u << 30);// addr[56:32] | type=2
    i32x8 g1;
    g1[0] = (2 << 16);           // data_size=2 (4 bytes); no multicast/iterate/pad
    g1[1] = (96 << 16);          // tensor_dim0[15:0] -> bits[63:48]
    g1[2] = (768 << 16);         // tensor_dim1[15:0] -> bits[95:80]
    g1[3] = (96 << 16);          // tile_dim0 = 96    -> bits[127:112]
    g1[4] = 768;                 // tile_dim1 = 768   -> bits[143:128]
    g1[5] = 96;                  // tensor_dim0_stride[31:0] = 96 elements
    g1[6] = 0;
    g1[7] = 0;
    i32x4 z4 = {0, 0, 0, 0};
#if __clang_major__ >= 23
    i32x8 z8 = {0, 0, 0, 0, 0, 0, 0, 0};
    __builtin_amdgcn_tensor_load_to_lds(g0, g1, z4, z4, z8, 0);
#else
    __builtin_amdgcn_tensor_load_to_lds(g0, g1, z4, z4, 0);
#endif
    __builtin_amdgcn_s_wait_tensorcnt(0);
  }
#else
  for (int i = threadIdx.x; i < EDIM * SEQLEN; i += blockDim.x)
    lds_x[i] = X[i];
#endif
  __syncthreads();
}

// ---------------------------------------------------------------------------
// Gather: seq[t][e] = emb1[x[t]][e]   (96x768, row-major)
//         xbuf[e][t] = emb2[x[t]][e]  (768x96, transposed for conv chain)
// ---------------------------------------------------------------------------
__global__ void gather_kernel(const int* __restrict__ x,
                              const float* __restrict__ emb1,
                              const float* __restrict__ emb2,
                              float* __restrict__ seq,
                              float* __restrict__ xbuf) {
  int idx = blockIdx.x * blockDim.x + threadIdx.x;
  if (idx >= SEQLEN * EDIM) return;
  int t = idx / EDIM;
  int e = idx - t * EDIM;
  size_t tok = (size_t)x[t];
  seq[(size_t)t * EDIM + e]  = emb1[tok * EDIM + e];
  xbuf[(size_t)e * SEQLEN + t] = emb2[tok * EDIM + e];
}

// ---------------------------------------------------------------------------
// WMMA GEMM: C(3072x96) = A(3072x768) @ Bt^T, Bt given row-major 96x768
// (so B[k][n] = Bt[n][k]).  One 16x16 output tile per wave.
// ---------------------------------------------------------------------------
__global__ void gemm_ih_wmma(const float* __restrict__ A,
                             const float* __restrict__ Bt,
                             float* __restrict__ C) {
  const int NT = SEQLEN / 16;                 // 6 N-tiles
  int wave = threadIdx.x >> 5;
  int tile = blockIdx.x * (blockDim.x >> 5) + wave;   // 192*6 = 1152 tiles
  int tm = (tile / NT) * 16;
  int tn = (tile % NT) * 16;
  int lane = threadIdx.x & 31;
  int half = lane >> 4;
  int l16  = lane & 15;

  v8f acc = {0.f,0.f,0.f,0.f,0.f,0.f,0.f,0.f};
  for (int k = 0; k < EDIM; k += 4) {
#ifdef HAVE_WMMA_F32
    int kk = k + 2 * half;
    v2f a, b;
    a.x = A[(size_t)(tm + l16) * EDIM + kk];
    a.y = A[(size_t)(tm + l16) * EDIM + kk + 1];
    b.x = Bt[(size_t)(tn + l16) * EDIM + kk];
    b.y = Bt[(size_t)(tn + l16) * EDIM + kk + 1];
    acc = wmma_f32(a, b, acc);
#else
    for (int r = 0; r < 8; ++r) {
      int m = tm + r + 8 * half;
      float s = 0.f;
      for (int k2 = 0; k2 < 4; ++k2)
        s += A[(size_t)m * EDIM + k + k2] * Bt[(size_t)(tn + l16) * EDIM + k + k2];
      acc[r] += s;
    }
#endif
  }
  for (int r = 0; r < 8; ++r) {
    int m = tm + r + 8 * half;
    C[(size_t)m * SEQLEN + tn + l16] = acc[r];
  }
}

// ---------------------------------------------------------------------------
// LSTM recurrence for one layer.  Gx = w_ih @ x_t precomputed (3072x96).
// Single workgroup of 768 threads (24 wave32s); h,c live in LDS.
// ---------------------------------------------------------------------------
__global__ void lstm_layer_kernel(const float* __restrict__ Gx,
                                  const float* __restrict__ w_hh,   // 3072x768
                                  const float* __restrict__ b_ih,   // 3072
                                  const float* __restrict__ b_hh,   // 3072
                                  const float* __restrict__ h0,     // 768
                                  const float* __restrict__ c0,     // 768
                                  float* __restrict__ ys,           // 96x768 or null
                                  float* __restrict__ h_final,      // 768
                                  float* __restrict__ c_final) {    // 768
  __shared__ float h[HDIM];
  __shared__ float c[HDIM];
  int j = threadIdx.x;
  h[j] = h0[j];
  c[j] = c0[j];
  __syncthreads();

  for (int t = 0; t < SEQLEN; ++t) {
    float g[4];
    #pragma unroll
    for (int gate = 0; gate < 4; ++gate) {
      int r = gate * HDIM + j;
      float acc = Gx[(size_t)r * SEQLEN + t] + b_ih[r] + b_hh[r];
      const float* wr = w_hh + (size_t)r * HDIM;
      #pragma unroll 4
      for (int k = 0; k < HDIM; ++k)
        acc = fmaf(wr[k], h[k], acc);
      g[gate] = acc;
    }
    float ig = sigmoidf_(g[0]);
    float fg = sigmoidf_(g[1]);
    float gg = tanhf(g[2]);
    float og = sigmoidf_(g[3]);
    float cn = fg * c[j] + ig * gg;
    float hn = og * tanhf(cn);
    __syncthreads();              // everyone done reading old h
    h[j] = hn;
    c[j] = cn;
    if (ys) ys[(size_t)t * HDIM + j] = hn;
    __syncthreads();              // new h visible before next step
  }
  h_final[j] = h[j];
  c_final[j] = c[j];
}

// ---------------------------------------------------------------------------
// One conv-chain step:  Y = relu(W0 @ X + W1 @ roll(X,-1,cols) + b)
// cw layout (E,E,2): cw[(m*768+k)*2 + {0,1}] = {W0,W1}[m][k].
// X staged into LDS via TDM; each wave computes a 16x32 tile (two
// accumulators share the weight fragments -> 4 WMMAs per 4 weight loads).
// ---------------------------------------------------------------------------
__global__ void conv_step_wmma(const float* __restrict__ cw,
                               const float* __restrict__ cb,   // 768
                               const float* __restrict__ X,    // 768x96
                               float* __restrict__ Y) {        // 768x96
  extern __shared__ float lds_x[];            // 768*96 floats = 288 KB
  stage_x_lds(X, lds_x);

  const int NP = SEQLEN / 32;                 // 3 N-pairs
  int wave = threadIdx.x >> 5;
  int tile = blockIdx.x * (blockDim.x >> 5) + wave;   // 48*3 = 144 tiles
  int tm = (tile / NP) * 16;
  int tn = (tile % NP) * 32;
  int lane = threadIdx.x & 31;
  int half = lane >> 4;
  int l16  = lane & 15;
  int n0  = tn + l16;                 // columns for acc0
  int n0s = (n0 + 1) % SEQLEN;        // rolled
  int n1  = n0 + 16;                  // columns for acc1
  int n1s = (n1 + 1) % SEQLEN;        // rolled

  v8f acc0 = {0.f,0.f,0.f,0.f,0.f,0.f,0.f,0.f};
  v8f acc1 = {0.f,0.f,0.f,0.f,0.f,0.f,0.f,0.f};
  for (int k = 0; k < EDIM; k += 4) {
    const float* wrow = cw + ((size_t)(tm + l16) * EDIM + k + 2 * half) * 2;
    __builtin_prefetch((const void*)(wrow + 64), 0, 0);   // weight stream prefetch
#ifdef HAVE_WMMA_F32
    int kk = k + 2 * half;
    v2f a0, a1, b00, b01, b10, b11;
    a0.x = wrow[0]; a1.x = wrow[1];
    a0.y = wrow[2]; a1.y = wrow[3];
    b00.x = lds_x[kk * SEQLEN + n0];        b00.y = lds_x[(kk + 1) * SEQLEN + n0];
    b01.x = lds_x[kk * SEQLEN + n0s];       b01.y = lds_x[(kk + 1) * SEQLEN + n0s];
    b10.x = lds_x[kk * SEQLEN + n1];        b10.y = lds_x[(kk + 1) * SEQLEN + n1];
    b11.x = lds_x[kk * SEQLEN + n1s];       b11.y = lds_x[(kk + 1) * SEQLEN + n1s];
    acc0 = wmma_f32(a0, b00, acc0);
    acc0 = wmma_f32(a1, b01, acc0);
    acc1 = wmma_f32(a0, b10, acc1);
    acc1 = wmma_f32(a1, b11, acc1);
#else
    for (int r = 0; r < 8; ++r) {
      int m = tm + r + 8 * half;
      float s0 = 0.f, s1 = 0.f;
      for (int k2 = 0; k2 < 4; ++k2) {
        const float* wp = cw + ((size_t)m * EDIM + k + k2) * 2;
        s0 += wp[0] * lds_x[(k + k2) * SEQLEN + n0];
        s0 += wp[1] * lds_x[(k + k2) * SEQLEN + n0s];
        s1 += wp[0] * lds_x[(k + k2) * SEQLEN + n1];
        s1 += wp[1] * lds_x[(k + k2) * SEQLEN + n1s];
      }
      acc0[r] += s0;
      acc1[r] += s1;
    }
#endif
  }
  for (int r = 0; r < 8; ++r) {
    int m = tm + r + 8 * half;
    float bb = cb[m];
    Y[(size_t)m * SEQLEN + n0] = fmaxf(acc0[r] + bb, 0.f);
    Y[(size_t)m * SEQLEN + n1] = fmaxf(acc1[r] + bb, 0.f);
  }
}

// ---------------------------------------------------------------------------
// Heads: out = feat @ pred_w^T + pred_b  with feat[h*2+l] = c[l][h]
//        out2 = sum_e m2w[e][0]*X[e][0] + m2w[e][1]*prev[e] + m2b
//        new_prev[e] = X[e][0]
// d_out layout: [0]=out | [1..1536]=h | [1537..3072]=c | [3073]=out2 |
//               [3074..3841]=new_prev
// ---------------------------------------------------------------------------
__global__ void finalize_kernel(const float* __restrict__ cbase,   // 2x768 (in d_out)
                                const float* __restrict__ pred_w,  // 1536
                                const float* __restrict__ pred_b,  // 1
                                const float* __restrict__ Xf,      // 768x96
                                const float* __restrict__ m2w,     // 768x2
                                const float* __restrict__ m2b,     // 1
                                const float* __restrict__ prev,    // 768
                                float* __restrict__ out) {
  __shared__ float red[256];
  int tid = threadIdx.x;

  float s1 = 0.f;
  for (int f = tid; f < 2 * HDIM; f += blockDim.x) {
    int hh = f >> 1, l = f & 1;
    s1 += cbase[l * HDIM + hh] * pred_w[f];
  }
  red[tid] = s1;
  __syncthreads();
  for (int off = 128; off > 0; off >>= 1) {
    if (tid < off) red[tid] += red[tid + off];
    __syncthreads();
  }
  if (tid == 0) out[0] = red[0] + pred_b[0];
  __syncthreads();

  float s2 = 0.f;
  for (int e = tid; e < EDIM; e += blockDim.x) {
    float m2o = Xf[(size_t)e * SEQLEN];   // column 0
    s2 += m2w[e * 2 + 0] * m2o + m2w[e * 2 + 1] * prev[e];
    out[3074 + e] = m2o;
  }
  red[tid] = s2;
  __syncthreads();
  for (int off = 128; off > 0; off >>= 1) {
    if (tid < off) red[tid] += red[tid + off];
    __syncthreads();
  }
  if (tid == 0) out[3073] = red[0] + m2b[0];
}

// ---------------------------------------------------------------------------
extern "C" void kernel_launch(void* const* d_in, const int* in_sizes, int n_in,
                              void* d_out, int out_size, void* d_ws, size_t ws_size,
                              hipStream_t stream) {
  const int*   x       = (const int*)d_in[0];
  const float* h_state = (const float*)d_in[1];
  const float* c_state = (const float*)d_in[2];
  const float* prev    = (const float*)d_in[3];
  const float* emb1    = (const float*)d_in[4];
  const float* emb2    = (const float*)d_in[5];
  const float* w_ih    = (const float*)d_in[6];
  const float* w_hh    = (const float*)d_in[7];
  const float* b_ih    = (const float*)d_in[8];
  const float* b_hh    = (const float*)d_in[9];
  const float* pred_w  = (const float*)d_in[10];
  const float* pred_b  = (const float*)d_in[11];
  const float* conv_w  = (const float*)d_in[12];
  const float* conv_b  = (const float*)d_in[13];
  const float* m2_w    = (const float*)d_in[14];
  const float* m2_b    = (const float*)d_in[15];
  float* out = (float*)d_out;

  float* ws   = (float*)d_ws;
  float* seq  = ws;                 // 96*768   = 73728
  float* ys0  = ws + 73728;         // 96*768   = 73728
  float* Gx   = ws + 147456;        // 3072*96  = 294912
  float* bufA = ws + 442368;        // 768*96   = 73728
  float* bufB = ws + 516096;        // 768*96   = 73728

  float* h_out = out + 1;
  float* c_out = out + 1537;

  // Embedding gather + transpose for conv chain
  gather_kernel<<<(SEQLEN * EDIM + 255) / 256, 256, 0, stream>>>(x, emb1, emb2, seq, bufA);

  // Layer 0: batched input GEMM (WMMA) + recurrence
  gemm_ih_wmma<<<288, 128, 0, stream>>>(w_ih, seq, Gx);
  lstm_layer_kernel<<<1, HDIM, 0, stream>>>(Gx, w_hh, b_ih, b_hh,
                                            h_state, c_state,
                                            ys0, h_out, c_out);

  // Layer 1: input GEMM over layer-0 outputs + recurrence
  gemm_ih_wmma<<<288, 128, 0, stream>>>(w_ih + (size_t)GDIM * EDIM, ys0, Gx);
  lstm_layer_kernel<<<1, HDIM, 0, stream>>>(Gx, w_hh + (size_t)GDIM * HDIM,
                                            b_ih + GDIM, b_hh + GDIM,
                                            h_state + HDIM, c_state + HDIM,
                                            nullptr, h_out + HDIM, c_out + HDIM);

  // Conv chain: 95 serial WMMA GEMM steps, ping-pong activation buffers.
  // Each workgroup TDM-stages the 288 KB activation matrix into LDS.
  const size_t lds_bytes = (size_t)EDIM * SEQLEN * sizeof(float);
  for (int l = 0; l < NCONV; ++l) {
    const float* Xl = (l & 1) ? bufB : bufA;
    float*       Yl = (l & 1) ? bufA : bufB;
    conv_step_wmma<<<36, 128, lds_bytes, stream>>>(conv_w + (size_t)l * EDIM * EDIM * 2,
                                                   conv_b + (size_t)l * EDIM,
                                                   Xl, Yl);
  }
  const float* Xfinal = (NCONV & 1) ? bufB : bufA;   // 95 odd -> bufB

  finalize_kernel<<<1, 256, 0, stream>>>(c_out, pred_w, pred_b,
                                         Xfinal, m2_w, m2_b, prev, out);
}